// LSTM_71897752535315
// MI455X (gfx1250) — compile-verified
//
#include <hip/hip_runtime.h>
#include <hip/hip_bf16.h>

// ---------------------------------------------------------------------------
// Fused 2-layer LSTM (H=20) + dense head, FP32 WMMA (V_WMMA_F32_16X16X4_F32).
// One wave32 per workgroup owns a 16-row batch tile for the full T=1024 scan.
// Weights are pre-swizzled in LDS into per-lane WMMA fragments (v2f), so each
// WMMA operand is one ds_load_b64 into an even-aligned VGPR pair (no moves).
// h state is written by the gate math directly in A-fragment layout.
// ---------------------------------------------------------------------------

typedef __attribute__((ext_vector_type(2))) float v2f;
typedef __attribute__((ext_vector_type(4))) float v4f;
typedef __attribute__((ext_vector_type(8))) float v8f;

#define LH    20     // hidden
#define LH4   80     // 4*H
#define LF    32     // input features
#define LT    1024   // timesteps
#define LB    256    // batch
#define LOUT  10     // dense head

__device__ __forceinline__ float sigf(float x) { return 1.0f / (1.0f + __expf(-x)); }
__device__ __forceinline__ float tanhfast(float x) {
    const float e = __expf(2.0f * x);
    return (e - 1.0f) / (e + 1.0f);
}

#define WMMA4(c, a, b) \
    (c) = __builtin_amdgcn_wmma_f32_16x16x4_f32(false, (a), false, (b), (short)0, (c), false, false)

__global__ void __launch_bounds__(32)
lstm2_dense_fused(const float* __restrict__ X,
                  const float* __restrict__ Wi1, const float* __restrict__ Wh1,
                  const float* __restrict__ b1,
                  const float* __restrict__ Wi2, const float* __restrict__ Wh2,
                  const float* __restrict__ b2,
                  const float* __restrict__ Wd,  const float* __restrict__ bd,
                  float* __restrict__ out)
{
    // Weight matrices as per-lane WMMA B fragments: frag f = (n*KK + kk),
    // sW[f*32 + lane] = { W[4kk + 2*(lane/16) + 0][16n + lane%16],
    //                    W[4kk + 2*(lane/16) + 1][16n + lane%16] }
    __shared__ v2f sWi1f[40 * 32];             // 5 N-tiles x 8 K-steps
    __shared__ v2f sWh1f[25 * 32];             // 5 x 5
    __shared__ v2f sWi2f[25 * 32];
    __shared__ v2f sWh2f[25 * 32];
    __shared__ v2f sWdf [ 5 * 32];             // 1 x 5 (N padded 10->16)
    __shared__ float sb1[LH4], sb2[LH4], sbd[16];
    __shared__ v4f sZt4[LH4 * 4];              // z transposed: [col][row] as float4
    __shared__ v2f sH1f[5 * 32], sH2f[5 * 32]; // h state in A-fragment layout
    __shared__ float sC1[16 * LH], sC2[16 * LH];

    const int lane  = threadIdx.x;             // 0..31
    const int mrow  = lane & 15;
    const int hi    = lane >> 4;
    const int khalf = 2 * hi;
    const int b0    = blockIdx.x * 16;

    // ---------------- stage weights as WMMA fragments ----------------
    for (int f = 0; f < 40; ++f) {             // Wi1 [32][80]
        const int n = f >> 3, kk = f & 7;
        const int row = 4 * kk + khalf, col = 16 * n + mrow;
        v2f w; w.x = Wi1[row * LH4 + col]; w.y = Wi1[(row + 1) * LH4 + col];
        sWi1f[f * 32 + lane] = w;
    }
    for (int f = 0; f < 25; ++f) {             // Wh1 / Wi2 / Wh2 [20][80]
        const int n = f / 5, kk = f % 5;
        const int row = 4 * kk + khalf, col = 16 * n + mrow;
        v2f w;
        w.x = Wh1[row * LH4 + col]; w.y = Wh1[(row + 1) * LH4 + col]; sWh1f[f * 32 + lane] = w;
        w.x = Wi2[row * LH4 + col]; w.y = Wi2[(row + 1) * LH4 + col]; sWi2f[f * 32 + lane] = w;
        w.x = Wh2[row * LH4 + col]; w.y = Wh2[(row + 1) * LH4 + col]; sWh2f[f * 32 + lane] = w;
    }
    for (int kk = 0; kk < 5; ++kk) {           // Wd [20][10], zero-pad cols 10..15
        const int row = 4 * kk + khalf;
        v2f w;
        w.x = (mrow < LOUT) ? Wd[row * LOUT + mrow] : 0.0f;
        w.y = (mrow < LOUT) ? Wd[(row + 1) * LOUT + mrow] : 0.0f;
        sWdf[kk * 32 + lane] = w;
    }
    for (int i = lane; i < LH4; i += 32) { sb1[i] = b1[i]; sb2[i] = b2[i]; }
    for (int i = lane; i < 16; i += 32) sbd[i] = (i < LOUT) ? bd[i] : 0.0f;
    {
        v2f zz; zz.x = 0.0f; zz.y = 0.0f;
        for (int i = lane; i < 5 * 32; i += 32) { sH1f[i] = zz; sH2f[i] = zz; }
        for (int i = lane; i < 16 * LH; i += 32) { sC1[i] = 0.0f; sC2[i] = 0.0f; }
    }
    __syncthreads();

    const float* zp = (const float*)sZt4;

    for (int t = 0; t < LT; ++t) {
        // ---- x_t A-fragments straight from global (b64 per lane per K-step)
        const float* xrow = X + ((size_t)(b0 + mrow) * LT + (size_t)t) * LF;
        v2f ax[8];
        #pragma unroll
        for (int kk = 0; kk < 8; ++kk) {
            ax[kk].x = xrow[4 * kk + khalf];
            ax[kk].y = xrow[4 * kk + khalf + 1];
        }
        if (t + 1 < LT) __builtin_prefetch(xrow + LF, 0, 3);   // global_prefetch_b8

        // ================ layer 1: z = x@Wi1 + h1@Wh1 (bias in gates) =======
        v8f zt[5];
        #pragma unroll
        for (int n = 0; n < 5; ++n) {
            v8f c = {};
            #pragma unroll
            for (int kk = 0; kk < 8; ++kk)
                WMMA4(c, ax[kk], sWi1f[(n * 8 + kk) * 32 + lane]);
            #pragma unroll
            for (int kk = 0; kk < 5; ++kk)
                WMMA4(c, sH1f[kk * 32 + lane], sWh1f[(n * 5 + kk) * 32 + lane]);
            zt[n] = c;
        }
        #pragma unroll
        for (int n = 0; n < 5; ++n) {          // transposed spill: 2x b128 per tile
            const int zcol = 16 * n + mrow;
            v4f lo, hiq;
            lo.x  = zt[n][0]; lo.y  = zt[n][1]; lo.z  = zt[n][2]; lo.w  = zt[n][3];
            hiq.x = zt[n][4]; hiq.y = zt[n][5]; hiq.z = zt[n][6]; hiq.w = zt[n][7];
            sZt4[zcol * 4 + khalf]     = lo;
            sZt4[zcol * 4 + khalf + 1] = hiq;
        }
        __syncthreads();

        // gate math layer 1 (flax order i,f,g,o); writes h in A-frag layout
        #pragma unroll
        for (int i = 0; i < 10; ++i) {
            const int idx = lane + 32 * i;     // = m*20 + j
            const int m = idx / LH, j = idx % LH;
            const float zi = zp[ j            * 16 + m] + sb1[j];
            const float zf = zp[(LH     + j)  * 16 + m] + sb1[LH + j];
            const float zg = zp[(2 * LH + j)  * 16 + m] + sb1[2 * LH + j];
            const float zo = zp[(3 * LH + j)  * 16 + m] + sb1[3 * LH + j];
            const float cc = sigf(zf) * sC1[idx] + sigf(zi) * tanhfast(zg);
            sC1[idx] = cc;
            const float hh = sigf(zo) * tanhfast(cc);
            const int kk = j >> 2, off = j & 3;
            ((float*)sH1f)[((kk * 32) + (m + 16 * (off >> 1))) * 2 + (off & 1)] = hh;
        }
        __syncthreads();

        // ================ layer 2: z = h1@Wi2 + h2@Wh2 ======================
        #pragma unroll
        for (int n = 0; n < 5; ++n) {
            v8f c = {};
            #pragma unroll
            for (int kk = 0; kk < 5; ++kk)
                WMMA4(c, sH1f[kk * 32 + lane], sWi2f[(n * 5 + kk) * 32 + lane]);
            #pragma unroll
            for (int kk = 0; kk < 5; ++kk)
                WMMA4(c, sH2f[kk * 32 + lane], sWh2f[(n * 5 + kk) * 32 + lane]);
            zt[n] = c;
        }
        #pragma unroll
        for (int n = 0; n < 5; ++n) {
            const int zcol = 16 * n + mrow;
            v4f lo, hiq;
            lo.x  = zt[n][0]; lo.y  = zt[n][1]; lo.z  = zt[n][2]; lo.w  = zt[n][3];
            hiq.x = zt[n][4]; hiq.y = zt[n][5]; hiq.z = zt[n][6]; hiq.w = zt[n][7];
            sZt4[zcol * 4 + khalf]     = lo;
            sZt4[zcol * 4 + khalf + 1] = hiq;
        }
        __syncthreads();

        #pragma unroll
        for (int i = 0; i < 10; ++i) {
            const int idx = lane + 32 * i;
            const int m = idx / LH, j = idx % LH;
            const float zi = zp[ j            * 16 + m] + sb2[j];
            const float zf = zp[(LH     + j)  * 16 + m] + sb2[LH + j];
            const float zg = zp[(2 * LH + j)  * 16 + m] + sb2[2 * LH + j];
            const float zo = zp[(3 * LH + j)  * 16 + m] + sb2[3 * LH + j];
            const float cc = sigf(zf) * sC2[idx] + sigf(zi) * tanhfast(zg);
            sC2[idx] = cc;
            const float hh = sigf(zo) * tanhfast(cc);
            const int kk = j >> 2, off = j & 3;
            ((float*)sH2f)[((kk * 32) + (m + 16 * (off >> 1))) * 2 + (off & 1)] = hh;
        }
        __syncthreads();

        // ================ dense head: out_t = h2 @ Wd + bd ==================
        {
            v8f c = {};
            #pragma unroll
            for (int kk = 0; kk < 5; ++kk)
                WMMA4(c, sH2f[kk * 32 + lane], sWdf[kk * 32 + lane]);
            if (mrow < LOUT) {
                const float bb = sbd[mrow];
                #pragma unroll
                for (int r = 0; r < 8; ++r) {
                    const int m = r + 8 * hi;
                    out[((size_t)(b0 + m) * LT + (size_t)t) * LOUT + mrow] = c[r] + bb;
                }
            }
        }
        __syncthreads();
    }
}

extern "C" void kernel_launch(void* const* d_in, const int* in_sizes, int n_in,
                              void* d_out, int out_size, void* d_ws, size_t ws_size,
                              hipStream_t stream) {
    (void)in_sizes; (void)n_in; (void)out_size; (void)d_ws; (void)ws_size;
    const float* X   = (const float*)d_in[0];
    const float* Wi1 = (const float*)d_in[1];
    const float* Wh1 = (const float*)d_in[2];
    const float* b1  = (const float*)d_in[3];
    const float* Wi2 = (const float*)d_in[4];
    const float* Wh2 = (const float*)d_in[5];
    const float* b2  = (const float*)d_in[6];
    const float* Wd  = (const float*)d_in[7];
    const float* bd  = (const float*)d_in[8];
    float* out = (float*)d_out;

    dim3 grid(LB / 16);   // 16 workgroups, one wave32 each
    dim3 block(32);
    lstm2_dense_fused<<<grid, block, 0, stream>>>(X, Wi1, Wh1, b1, Wi2, Wh2, b2, Wd, bd, out);
}